// GameCell_39565238731289
// MI455X (gfx1250) — compile-verified
//
#include <hip/hip_runtime.h>
#include <hip/hip_bf16.h>

// ---------------------------------------------------------------------------
// GameCell GRU-style update on MI455X (gfx1250).
// Bandwidth-bound: 3 x (4096x4096) fp32 mat-vecs = 201 MB/call -> ~8.6us floor
// at 23.3 TB/s. Uses V_WMMA_F32_16X16X4_F32 (fp32 in/out, no precision loss)
// with split-K (16 waves per 16-row tile) for memory-level parallelism.
// Phase 1 (Az, Ar) merged into one launch via grid.y -> 8192 concurrent waves.
// ---------------------------------------------------------------------------

#define STATES 2048
#define TEAMS  32
#define U_DIM  64
#define S2     4096            // 2*STATES
#define KSEGS  16              // waves per row-tile (split-K)
#define KSEG_LEN (S2 / KSEGS)  // 256
#define GEMV_BLOCK (KSEGS * 32) // 512 threads

typedef float v2f __attribute__((ext_vector_type(2)));
typedef float v8f __attribute__((ext_vector_type(8)));

// ---------------------------------------------------------------------------
// Kernel 0: copy state -> out, gather x = state[team_idx].reshape(4096)
// ---------------------------------------------------------------------------
__global__ __launch_bounds__(256) void prep_kernel(
    const float* __restrict__ state, const int* __restrict__ team_idx,
    float* __restrict__ out, float* __restrict__ xbuf) {
  const int i = blockIdx.x * blockDim.x + threadIdx.x;   // 0..65535
  out[i] = state[i];
  if (i < S2) {
    const int team = team_idx[i >> 11];                  // i / STATES
    xbuf[i] = state[team * STATES + (i & (STATES - 1))];
  }
}

// ---------------------------------------------------------------------------
// GEMV via WMMA f32 16x16x4.
//   grid.x = S2/16 = 256 row tiles, grid.y selects matrix {0,1},
//   block = 512 threads = 16 waves (split-K segments of 256).
//   Partial 16-row sums reduced in LDS, one clean store per tile (no atomics).
// A-matrix lane layout (ISA 7.12.2, 32-bit A 16x4):
//   lane L<16 : row L,    v0=K0, v1=K1
//   lane L>=16: row L-16, v0=K2, v1=K3
// B is the x chunk replicated across all 16 columns -> every column of D is
// the same partial dot product, so lanes 0 / 16 hold all 16 row results.
// ---------------------------------------------------------------------------
__global__ __launch_bounds__(GEMV_BLOCK) void gemv_wmma_kernel(
    const float* __restrict__ A0, const float* __restrict__ A1,
    const float* __restrict__ xin,
    float* __restrict__ y0, float* __restrict__ y1) {
  __shared__ float red[KSEGS * 16];

  const float* __restrict__ A    = (blockIdx.y == 0) ? A0 : A1;
  float* __restrict__       yout = (blockIdx.y == 0) ? y0 : y1;

  const int lane = threadIdx.x & 31;
  const int wave = threadIdx.x >> 5;                 // K segment 0..15
  const int tile = blockIdx.x;                       // 16-row tile
  const int row  = tile * 16 + (lane & 15);
  const int koff = (lane >> 4) << 1;                 // 0 (lanes<16) or 2

  const float* __restrict__ arow = A + (size_t)row * S2 + koff;
  const float* __restrict__ xoff = xin + koff;

  v8f acc0 = {0.f, 0.f, 0.f, 0.f, 0.f, 0.f, 0.f, 0.f};
  v8f acc1 = {0.f, 0.f, 0.f, 0.f, 0.f, 0.f, 0.f, 0.f};

  const int k0 = wave * KSEG_LEN;
#pragma unroll 4
  for (int k = k0; k < k0 + KSEG_LEN; k += 8) {
    v2f a0 = *(const v2f*)(arow + k);
    v2f b0 = *(const v2f*)(xoff + k);
    v2f a1 = *(const v2f*)(arow + k + 4);
    v2f b1 = *(const v2f*)(xoff + k + 4);
    // D = A(16x4) * B(4x16) + C   (fp32 throughout)
    acc0 = __builtin_amdgcn_wmma_f32_16x16x4_f32(
        false, a0, false, b0, (short)0, acc0, false, false);
    acc1 = __builtin_amdgcn_wmma_f32_16x16x4_f32(
        false, a1, false, b1, (short)0, acc1, false, false);
  }
  v8f acc = acc0 + acc1;

  // lane 0 holds rows 0..7 (all columns identical), lane 16 holds rows 8..15
  if (lane == 0 || lane == 16) {
    const int base = wave * 16 + ((lane >> 4) << 3);
#pragma unroll
    for (int r = 0; r < 8; ++r) red[base + r] = acc[r];
  }
  __syncthreads();

  if (threadIdx.x < 16) {
    float s = 0.f;
#pragma unroll
    for (int w = 0; w < KSEGS; ++w) s += red[w * 16 + threadIdx.x];
    yout[tile * 16 + threadIdx.x] = s;
  }
}

// ---------------------------------------------------------------------------
// Finalize phase 1: z = sigmoid(accZ + Bz@u + dz), r = sigmoid(accR + Br@u - dr)
// store z and r*x for phase 2.
// ---------------------------------------------------------------------------
__global__ __launch_bounds__(256) void finalize_zr_kernel(
    const float* __restrict__ accZ, const float* __restrict__ accR,
    const float* __restrict__ Bz, const float* __restrict__ Br,
    const float* __restrict__ u, const float* __restrict__ dz,
    const float* __restrict__ dr, const float* __restrict__ x,
    float* __restrict__ zbuf, float* __restrict__ rxbuf) {
  const int i = blockIdx.x * blockDim.x + threadIdx.x;   // 0..4095
  const float4* __restrict__ bz = (const float4*)(Bz + (size_t)i * U_DIM);
  const float4* __restrict__ br = (const float4*)(Br + (size_t)i * U_DIM);
  const float4* __restrict__ u4 = (const float4*)u;
  float duz = 0.f, dur = 0.f;
#pragma unroll
  for (int j = 0; j < U_DIM / 4; ++j) {
    const float4 a = bz[j], b = br[j], c = u4[j];
    duz += a.x * c.x + a.y * c.y + a.z * c.z + a.w * c.w;
    dur += b.x * c.x + b.y * c.y + b.z * c.z + b.w * c.w;
  }
  const float zp = accZ[i] + duz + dz[i];
  const float rp = accR[i] + dur - dr[i];
  const float z = 1.f / (1.f + __expf(-zp));
  const float r = 1.f / (1.f + __expf(-rp));
  zbuf[i]  = z;
  rxbuf[i] = r * x[i];
}

// ---------------------------------------------------------------------------
// Finalize phase 2: m = tanh(accM + Bm@u + dm); x_new = z*x + (1-z)*m;
// scatter-add dx into out (atomic -> matches JAX .at[].add for dup indices).
// ---------------------------------------------------------------------------
__global__ __launch_bounds__(256) void finalize_m_kernel(
    const float* __restrict__ accM, const float* __restrict__ Bm,
    const float* __restrict__ u, const float* __restrict__ dm,
    const float* __restrict__ x, const float* __restrict__ zbuf,
    const int* __restrict__ team_idx, float* __restrict__ out) {
  const int i = blockIdx.x * blockDim.x + threadIdx.x;   // 0..4095
  const float4* __restrict__ bm = (const float4*)(Bm + (size_t)i * U_DIM);
  const float4* __restrict__ u4 = (const float4*)u;
  float dum = 0.f;
#pragma unroll
  for (int j = 0; j < U_DIM / 4; ++j) {
    const float4 a = bm[j], c = u4[j];
    dum += a.x * c.x + a.y * c.y + a.z * c.z + a.w * c.w;
  }
  const float m  = tanhf(accM[i] + dum + dm[i]);
  const float xi = x[i];
  const float z  = zbuf[i];
  const float dx = (z * xi + (1.f - z) * m) - xi;
  const int team = team_idx[i >> 11];
  atomicAdd(out + team * STATES + (i & (STATES - 1)), dx);
}

// ---------------------------------------------------------------------------
// Host-side launcher (graph-capture safe: only kernel launches on `stream`).
// Input order: team_idx, u, state, Bz, Br, Bm, Az, Ar, Am, dz, dr, dm
// ---------------------------------------------------------------------------
extern "C" void kernel_launch(void* const* d_in, const int* in_sizes, int n_in,
                              void* d_out, int out_size, void* d_ws, size_t ws_size,
                              hipStream_t stream) {
  (void)in_sizes; (void)n_in; (void)out_size; (void)ws_size;

  const int*   team_idx = (const int*)  d_in[0];
  const float* u        = (const float*)d_in[1];
  const float* state    = (const float*)d_in[2];
  const float* Bz       = (const float*)d_in[3];
  const float* Br       = (const float*)d_in[4];
  const float* Bm       = (const float*)d_in[5];
  const float* Az       = (const float*)d_in[6];
  const float* Ar       = (const float*)d_in[7];
  const float* Am       = (const float*)d_in[8];
  const float* dz       = (const float*)d_in[9];
  const float* dr       = (const float*)d_in[10];
  const float* dm       = (const float*)d_in[11];
  float* out = (float*)d_out;

  // Workspace layout (floats): x | z | rx | accZ | accR | accM  (96 KB)
  float* ws   = (float*)d_ws;
  float* xb   = ws + 0 * S2;
  float* zb   = ws + 1 * S2;
  float* rxb  = ws + 2 * S2;
  float* accZ = ws + 3 * S2;
  float* accR = ws + 4 * S2;
  float* accM = ws + 5 * S2;

  // K0: copy state -> out, gather x
  prep_kernel<<<TEAMS * STATES / 256, 256, 0, stream>>>(state, team_idx, out, xb);

  // K1: merged split-K WMMA GEMVs for z and r gates (grid.y selects Az/Ar)
  gemv_wmma_kernel<<<dim3(S2 / 16, 2), GEMV_BLOCK, 0, stream>>>(
      Az, Ar, xb, accZ, accR);

  // K2: activations for z, r; build r*x
  finalize_zr_kernel<<<S2 / 256, 256, 0, stream>>>(accZ, accR, Bz, Br, u, dz, dr,
                                                   xb, zb, rxb);

  // K3: WMMA GEMV for the m gate on r*x
  gemv_wmma_kernel<<<dim3(S2 / 16, 1), GEMV_BLOCK, 0, stream>>>(
      Am, Am, rxb, accM, accM);

  // K4: m activation, GRU blend, scatter-add delta into out
  finalize_m_kernel<<<S2 / 256, 256, 0, stream>>>(accM, Bm, u, dm, xb, zb,
                                                  team_idx, out);
}